// Attention_70927089926531
// MI455X (gfx1250) — compile-verified
//
#include <hip/hip_runtime.h>
#include <stdint.h>

#define S_LEN 512
#define DDIM  512
#define WAVES 8
#define STEPS (S_LEN / WAVES)   // 64 s-steps per wave

typedef float v2f __attribute__((ext_vector_type(2)));
typedef float v8f __attribute__((ext_vector_type(8)));

// ---------------------------------------------------------------------------
// Kernel 1: hb[s] = dot(hidden[s,:], W[512:1024]) + b   (512 values into d_ws)
// ---------------------------------------------------------------------------
__global__ __launch_bounds__(256) void hdot_kernel(
    const float* __restrict__ hidden, const float* __restrict__ W,
    const float* __restrict__ b, float* __restrict__ hb) {
  const int lane = threadIdx.x & 31;
  const int wave = threadIdx.x >> 5;
  const int s = blockIdx.x * WAVES + wave;
  const float4* hrow = (const float4*)(hidden + (size_t)s * DDIM);
  const float4* wh   = (const float4*)(W + DDIM);
  float sum = 0.f;
#pragma unroll
  for (int k = 0; k < 4; ++k) {
    float4 hv = hrow[lane * 4 + k];
    float4 wv = wh[lane * 4 + k];
    sum += hv.x * wv.x + hv.y * wv.y + hv.z * wv.z + hv.w * wv.w;
  }
#pragma unroll
  for (int off = 16; off >= 1; off >>= 1)
    sum += __shfl_xor(sum, off, 32);
  if (lane == 0) hb[s] = sum + b[0];
}

// ---------------------------------------------------------------------------
// Async 2KB row copy: 32 lanes x 16B x 4 instructions. The immediate offset
// applies to both the global and LDS addresses (CDNA5 async semantics).
// NT hint: the 512MB tensor is streamed exactly once -> keep L2 clean.
// ---------------------------------------------------------------------------
__device__ __forceinline__ void async_row(uint32_t lds_lane_addr, const void* g_lane_addr) {
  asm volatile(
      "global_load_async_to_lds_b128 %0, %1, off th:TH_LOAD_NT\n\t"
      "global_load_async_to_lds_b128 %0, %1, off offset:512 th:TH_LOAD_NT\n\t"
      "global_load_async_to_lds_b128 %0, %1, off offset:1024 th:TH_LOAD_NT\n\t"
      "global_load_async_to_lds_b128 %0, %1, off offset:1536 th:TH_LOAD_NT"
      :: "v"(lds_lane_addr), "v"(g_lane_addr) : "memory");
}

// ---------------------------------------------------------------------------
// Kernel 2: fused single-pass energy + online softmax + weighted sum.
// One block per t-column; wave w handles s = w, w+8, ...
// ---------------------------------------------------------------------------
__global__ __launch_bounds__(256) void attn_fused_kernel(
    const float* __restrict__ enc, const float* __restrict__ W,
    const float* __restrict__ hb, float* __restrict__ out) {
  __shared__ float stage[WAVES][2][DDIM];   // 32 KB async staging
  __shared__ float mergec[WAVES][DDIM];     // 16 KB merge area
  __shared__ float mw[WAVES], lw[WAVES];

  const int t    = blockIdx.x;
  const int lane = threadIdx.x & 31;
  const int wave = threadIdx.x >> 5;

  // Per-lane slice of We (16 floats) kept in VGPRs for the whole kernel.
  float4 w0, w1, w2, w3;
  {
    const float4* wv = (const float4*)W;
    w0 = wv[lane * 4 + 0];
    w1 = wv[lane * 4 + 1];
    w2 = wv[lane * 4 + 2];
    w3 = wv[lane * 4 + 3];
  }

  // LDS byte addresses for this lane's 16B staging slot in each buffer
  // (low 32 bits of the generic LDS address = LDS byte offset).
  const uint32_t lds_a[2] = {
      (uint32_t)(uintptr_t)((const uint8_t*)&stage[wave][0][0] + lane * 16),
      (uint32_t)(uintptr_t)((const uint8_t*)&stage[wave][1][0] + lane * 16)};

  // Context accumulators (lane owns d = lane*16 .. lane*16+15), online softmax state.
  float4 c0 = {0.f, 0.f, 0.f, 0.f}, c1 = c0, c2 = c0, c3 = c0;
  float m = -__builtin_inff();
  float l = 0.f;

  const size_t row_stride = (size_t)WAVES * S_LEN * DDIM;  // advance s by 8
  const float* g = enc + ((size_t)wave * S_LEN + t) * DDIM;

  // Prime the pipeline: prefetch step 0 into buffer 0.
  async_row(lds_a[0], (const uint8_t*)g + lane * 16);

#pragma unroll 2
  for (int step = 0; step < STEPS; ++step) {
    const int buf = step & 1;
    const float* gn = g + row_stride;
    if (step + 1 < STEPS) {
      async_row(lds_a[buf ^ 1], (const uint8_t*)gn + lane * 16);
      asm volatile("s_wait_asynccnt 0x4" ::: "memory");  // this step's 4 copies done
    } else {
      asm volatile("s_wait_asynccnt 0x0" ::: "memory");
    }

    // Lane reads its 16 contiguous floats of the staged row.
    const float4* sp = (const float4*)&stage[wave][buf][lane * 16];
    float4 v0 = sp[0], v1 = sp[1], v2 = sp[2], v3 = sp[3];

    // Partial dot(enc_row, We) then wave32 reduction.
    float e = v0.x * w0.x + v0.y * w0.y + v0.z * w0.z + v0.w * w0.w
            + v1.x * w1.x + v1.y * w1.y + v1.z * w1.z + v1.w * w1.w
            + v2.x * w2.x + v2.y * w2.y + v2.z * w2.z + v2.w * w2.w
            + v3.x * w3.x + v3.y * w3.y + v3.z * w3.z + v3.w * w3.w;
#pragma unroll
    for (int off = 16; off >= 1; off >>= 1)
      e += __shfl_xor(e, off, 32);

    e += hb[wave + step * WAVES];                    // + hdot[s] + b
    float en = 1.f - 2.f / (__expf(2.f * e) + 1.f);  // tanh(e), stable

    // Online softmax update.
    float mn    = fmaxf(m, en);
    float alpha = __expf(m - mn);
    float p     = __expf(en - mn);
    l = l * alpha + p;
#define UPD(cc, vv)                       \
    cc.x = cc.x * alpha + p * vv.x;       \
    cc.y = cc.y * alpha + p * vv.y;       \
    cc.z = cc.z * alpha + p * vv.z;       \
    cc.w = cc.w * alpha + p * vv.w
    UPD(c0, v0); UPD(c1, v1); UPD(c2, v2); UPD(c3, v3);
#undef UPD
    m = mn;
    g = gn;
  }

  // Publish per-wave partials.
  {
    float4* mc = (float4*)&mergec[wave][lane * 16];
    mc[0] = c0; mc[1] = c1; mc[2] = c2; mc[3] = c3;
    if (lane == 0) { mw[wave] = m; lw[wave] = l; }
  }
  __syncthreads();

  // Split-softmax merge scalars (all lanes compute them; 8 values).
  float M = mw[0];
#pragma unroll
  for (int w = 1; w < WAVES; ++w) M = fmaxf(M, mw[w]);
  float scale[WAVES];
  float L = 0.f;
#pragma unroll
  for (int w = 0; w < WAVES; ++w) {
    scale[w] = __expf(mw[w] - M);
    L += lw[w] * scale[w];
  }
  const float invL = 1.f / L;

  // WMMA merge: context[dchunk] = sum_{k=0..7} scale[k] * mergec[k][dchunk]
  // as chained V_WMMA_F32_16X16X4_F32. A(16x4) = scale[k] replicated down M
  // (lanes 0-15: K0/K1 in vgpr0/1; lanes 16-31: K2/K3). B(4x16) = mergec rows
  // (vgpr0 = K0 | K2 across lane halves, vgpr1 = K1 | K3). D rows are all
  // identical; row 0 lives in result element 0 of lanes 0-15.
  const bool hi = (lane >= 16);
  const int  n  = lane & 15;
  v8f acc[4];
#pragma unroll
  for (int ch = 0; ch < 4; ++ch) {
    const int dbase = wave * 64 + ch * 16;
    v8f c = {0.f, 0.f, 0.f, 0.f, 0.f, 0.f, 0.f, 0.f};
#pragma unroll
    for (int kk = 0; kk < 2; ++kk) {
      const int k0 = kk * 4;
      v2f a, bm;
      a.x  = hi ? scale[k0 + 2] : scale[k0 + 0];
      a.y  = hi ? scale[k0 + 3] : scale[k0 + 1];
      bm.x = hi ? mergec[k0 + 2][dbase + n] : mergec[k0 + 0][dbase + n];
      bm.y = hi ? mergec[k0 + 3][dbase + n] : mergec[k0 + 1][dbase + n];
      c = __builtin_amdgcn_wmma_f32_16x16x4_f32(
          /*neg_a=*/false, a, /*neg_b=*/false, bm,
          /*c_mod=*/(short)0, c, /*reuse_a=*/false, /*reuse_b=*/false);
    }
    acc[ch] = c;
  }
  if (lane < 16) {
#pragma unroll
    for (int ch = 0; ch < 4; ++ch)
      out[(size_t)t * DDIM + wave * 64 + ch * 16 + n] = acc[ch][0] * invL;
  }
}

// ---------------------------------------------------------------------------
extern "C" void kernel_launch(void* const* d_in, const int* in_sizes, int n_in,
                              void* d_out, int out_size, void* d_ws, size_t ws_size,
                              hipStream_t stream) {
  (void)in_sizes; (void)n_in; (void)out_size; (void)ws_size;
  const float* enc    = (const float*)d_in[0];  // (512,512,512) f32
  const float* hidden = (const float*)d_in[1];  // (512,512) f32
  const float* W      = (const float*)d_in[2];  // (1024,1) f32
  const float* b      = (const float*)d_in[3];  // (1,) f32
  float* out = (float*)d_out;                   // (1,512,512) f32
  float* hb  = (float*)d_ws;                    // 512 floats scratch

  hdot_kernel<<<S_LEN / WAVES, 256, 0, stream>>>(hidden, W, b, hb);
  attn_fused_kernel<<<S_LEN, 256, 0, stream>>>(enc, W, hb, out);
}